// CustomHyperSemanticMessagePassing_18622978195701
// MI455X (gfx1250) — compile-verified
//
#include <hip/hip_runtime.h>
#include <math.h>

typedef __attribute__((ext_vector_type(2))) float v2f;
typedef __attribute__((ext_vector_type(8))) float v8f;

#define D_   128   // node feature / out dim
#define ED_  64    // edge feature dim
#define H_   8     // heads
#define DH_  16    // head dim
#define K_   4     // nodes per hyperedge

static __device__ __forceinline__ v8f wmma4f32(v2f a, v2f b, v8f c) {
  // V_WMMA_F32_16X16X4_F32 : full fp32 matrix pipe on gfx1250
  return __builtin_amdgcn_wmma_f32_16x16x4_f32(false, a, false, b, (short)0, c, false, false);
}

// order-preserving float<->uint mapping for atomic segment-max
static __device__ __forceinline__ unsigned encodeOrd(float f) {
  unsigned u = __float_as_uint(f);
  return (u & 0x80000000u) ? ~u : (u | 0x80000000u);
}
static __device__ __forceinline__ float decodeOrd(unsigned u) {
  return (u & 0x80000000u) ? __uint_as_float(u & 0x7fffffffu) : __uint_as_float(~u);
}

// ---------------------------------------------------------------------------
// Out[M x 128] = act( (Arow @ W[Kd x 128] + bias) * scale )
//   GATHER=false: Arow = A[row]               (dense A, lda = Kd)
//   GATHER=true : Arow = A[u] + We[e]         (key vectors; row s = e*4+j,
//                                              u = pair_u[e*16+j], Kd == 128)
//
// Block = (blockDim.x/64) row-tiles x 2 column-halves; one wave owns a
// 16x64 output tile. W is staged in LDS *transposed* (stride Kd+2) so each
// lane's B fragment (W[ka][n], W[ka+1][n]) is one aligned ds_load_b64.
// A-fragment per ISA layout: lane m=lane&15 -> row m, lane>>4 selects K pair.
// ---------------------------------------------------------------------------
template<bool GATHER>
__global__ void gemm_wmma_kernel(const float* __restrict__ A,
                                 const float* __restrict__ Wed,   // GATHER: We matrix
                                 const int* __restrict__ pair_u,  // GATHER: slot->node
                                 const float* __restrict__ W,
                                 const float* __restrict__ bias,
                                 float* __restrict__ Out,
                                 int Kd, float scale, int relu) {
  extern __shared__ float Wt[];               // [128][Kd+2] transposed weights
  const int stride = Kd + 2;

  // stage W transposed: global row-major reads are coalesced
  for (int i = threadIdx.x; i < Kd * D_; i += blockDim.x) {
    const int r = i >> 7;                     // k index
    const int c = i & 127;                    // n index
    Wt[c * stride + r] = W[i];
  }
  __syncthreads();

  const int lane = threadIdx.x & 31;
  const int wave = threadIdx.x >> 5;
  const int m    = lane & 15;
  const int kh   = lane >> 4;                 // 0/1 selects K pair
  const int colBase = (wave & 1) * 64;
  const int rowTile = blockIdx.x * (blockDim.x >> 6) + (wave >> 1);
  const int row0 = rowTile * 16;

  const float* __restrict__ a0;
  const float* __restrict__ a1 = nullptr;
  if (GATHER) {
    const int s = row0 + m;                   // edge-slot row
    const int e = s >> 2;
    const int j = s & 3;
    const int u = pair_u[e * 16 + j];
    a0 = A   + (size_t)u * D_;                // Wh[u]
    a1 = Wed + (size_t)e * D_;                // We[e]
  } else {
    a0 = A + (size_t)(row0 + m) * Kd;
  }

  const float* __restrict__ wc0 = Wt + (size_t)(colBase + m) * stride;
  const float* __restrict__ wc1 = wc0 + 16 * stride;
  const float* __restrict__ wc2 = wc0 + 32 * stride;
  const float* __restrict__ wc3 = wc0 + 48 * stride;

  v8f acc0 = {}, acc1 = {}, acc2 = {}, acc3 = {};
#pragma unroll 4
  for (int k0 = 0; k0 < Kd; k0 += 4) {
    const int ka = k0 + 2 * kh;
    v2f a = *(const v2f*)(a0 + ka);
    if (GATHER) a += *(const v2f*)(a1 + ka);
    v2f b0 = *(const v2f*)(wc0 + ka);         // ds_load_b64 each
    v2f b1 = *(const v2f*)(wc1 + ka);
    v2f b2 = *(const v2f*)(wc2 + ka);
    v2f b3 = *(const v2f*)(wc3 + ka);
    acc0 = wmma4f32(a, b0, acc0);
    acc1 = wmma4f32(a, b1, acc1);
    acc2 = wmma4f32(a, b2, acc2);
    acc3 = wmma4f32(a, b3, acc3);
  }

#pragma unroll
  for (int r = 0; r < 8; ++r) {
    const int outRow = row0 + r + 8 * kh;     // C/D layout: VGPR r -> M=r (+8 upper half)
    float vals[4] = {acc0[r], acc1[r], acc2[r], acc3[r]};
#pragma unroll
    for (int t = 0; t < 4; ++t) {
      const int col = colBase + 16 * t + m;
      float v = vals[t];
      if (bias) v += bias[col];
      v *= scale;
      if (relu) v = fmaxf(v, 0.0f);
      Out[(size_t)outRow * D_ + col] = v;
    }
  }
}

// ---------------------------------------------------------------------------
// Per owner-group g = e*4 + i (4 pairs j=0..3, same owner v):
// lane = (j,h): score = <Q[v,h], Kv[e*4+j,h]>, stash, atomic segment-max.
// ---------------------------------------------------------------------------
__global__ void scores_kernel(const float* __restrict__ Q,
                              const float* __restrict__ Kv,
                              const int* __restrict__ owners,
                              float* __restrict__ sc,
                              unsigned* __restrict__ mEnc,
                              int G) {
  const int g = blockIdx.x * (blockDim.x >> 5) + (threadIdx.x >> 5);
  if (g >= G) return;
  const int lane = threadIdx.x & 31;
  const int j = lane >> 3;
  const int h = lane & 7;
  const int v = owners[4 * g];                // pair index 4g = e*16 + i*4
  const int skey = (g >> 2) * 4 + j;          // e*4 + j
  const float4* __restrict__ q = (const float4*)(Q  + (size_t)v * D_ + h * DH_);
  const float4* __restrict__ k = (const float4*)(Kv + (size_t)skey * D_ + h * DH_);
  float s = 0.0f;
#pragma unroll
  for (int t = 0; t < 4; ++t) {
    float4 qa = q[t], ka = k[t];
    s = fmaf(qa.x, ka.x, s);
    s = fmaf(qa.y, ka.y, s);
    s = fmaf(qa.z, ka.z, s);
    s = fmaf(qa.w, ka.w, s);
  }
  sc[(size_t)g * 32 + lane] = s;
  float mx = s;
  mx = fmaxf(mx, __shfl_xor(mx, 8, 32));
  mx = fmaxf(mx, __shfl_xor(mx, 16, 32));
  if (j == 0) atomicMax(&mEnc[(size_t)v * H_ + h], encodeOrd(mx));
}

// exp(score - m) in place, atomic segment-sum into denom
__global__ void expsum_kernel(float* __restrict__ sc,
                              const unsigned* __restrict__ mEnc,
                              const int* __restrict__ owners,
                              float* __restrict__ denom,
                              int G) {
  const int g = blockIdx.x * (blockDim.x >> 5) + (threadIdx.x >> 5);
  if (g >= G) return;
  const int lane = threadIdx.x & 31;
  const int j = lane >> 3;
  const int h = lane & 7;
  const int v = owners[4 * g];
  const float mval = decodeOrd(mEnc[(size_t)v * H_ + h]);
  float e = expf(sc[(size_t)g * 32 + lane] - mval);
  sc[(size_t)g * 32 + lane] = e;
  float ssum = e;
  ssum += __shfl_xor(ssum, 8, 32);
  ssum += __shfl_xor(ssum, 16, 32);
  if (j == 0) atomicAdd(&denom[(size_t)v * H_ + h], ssum);
}

// ctx[v] += sum_j alpha[j,h] * V[u_j]  (per group, local j-reduction, 1 atomic/elem)
__global__ void ctx_kernel(const float* __restrict__ sc,
                           const float* __restrict__ denom,
                           const float* __restrict__ Vv,
                           const int* __restrict__ owners,
                           const int* __restrict__ pair_u,
                           float* __restrict__ ctx,
                           int G) {
  const int g = blockIdx.x * (blockDim.x >> 5) + (threadIdx.x >> 5);
  if (g >= G) return;
  const int lane = threadIdx.x & 31;
  const int v = owners[4 * g];
  const int e = g >> 2;
  const int u0 = pair_u[e * 16 + 0];
  const int u1 = pair_u[e * 16 + 1];
  const int u2 = pair_u[e * 16 + 2];
  const int u3 = pair_u[e * 16 + 3];
  const float* __restrict__ scg = sc + (size_t)g * 32;
#pragma unroll
  for (int r = 0; r < 4; ++r) {
    const int d = lane + 32 * r;
    const int h = d >> 4;
    const float dn = denom[(size_t)v * H_ + h];
    const float a0 = scg[0 * 8 + h];
    const float a1 = scg[1 * 8 + h];
    const float a2 = scg[2 * 8 + h];
    const float a3 = scg[3 * 8 + h];
    float acc = a0 * Vv[(size_t)u0 * D_ + d] + a1 * Vv[(size_t)u1 * D_ + d]
              + a2 * Vv[(size_t)u2 * D_ + d] + a3 * Vv[(size_t)u3 * D_ + d];
    atomicAdd(&ctx[(size_t)v * D_ + d], acc / dn);
  }
}

// ---------------------------------------------------------------------------
extern "C" void kernel_launch(void* const* d_in, const int* in_sizes, int n_in,
                              void* d_out, int out_size, void* d_ws, size_t ws_size,
                              hipStream_t stream) {
  const float* x         = (const float*)d_in[0];
  const float* edge_attr = (const float*)d_in[1];
  const float* w_lin     = (const float*)d_in[2];
  const float* w_e       = (const float*)d_in[3];
  const float* w_q       = (const float*)d_in[4];
  const float* b_q       = (const float*)d_in[5];
  const float* w_k       = (const float*)d_in[6];
  const float* b_k       = (const float*)d_in[7];
  const float* w_v       = (const float*)d_in[8];
  const float* b_v       = (const float*)d_in[9];
  const float* w_o       = (const float*)d_in[10];
  const float* b_o       = (const float*)d_in[11];
  const int*   owners    = (const int*)d_in[12];
  // d_in[13] = pair_e (implied by layout, unused)
  const int*   pair_u    = (const int*)d_in[14];

  const int N  = in_sizes[0] / D_;
  const int E  = in_sizes[1] / ED_;
  const int EK = E * K_;           // edge-slot rows (key vectors)
  const int G  = E * K_;           // owner groups (e, i)

  // workspace layout (fp32)
  float* ws   = (float*)d_ws;
  size_t o = 0;
  float* Wh   = ws + o; o += (size_t)N * D_;
  float* Q    = ws + o; o += (size_t)N * D_;
  float* Vv   = ws + o; o += (size_t)N * D_;
  float* Wem  = ws + o; o += (size_t)E * D_;
  float* Kv   = ws + o; o += (size_t)EK * D_;
  float* sc   = ws + o; o += (size_t)G * 32;
  float* ctx  = ws + o; o += (size_t)N * D_;          // ctx, denom, mEnc contiguous
  float* denom = ws + o; o += (size_t)N * H_;
  unsigned* mEnc = (unsigned*)(ws + o); o += (size_t)N * H_;

  const size_t lds128 = (size_t)D_ * (D_  + 2) * sizeof(float);   // 128x130 f32
  const size_t lds64  = (size_t)D_ * (ED_ + 2) * sizeof(float);   // 128x66  f32

  // Dense projections (WMMA f32 16x16x4), W staged transposed in LDS.
  // N/32 and EK/32 are exact; E/32 is not, so the E GEMM uses 1 row-tile/block.
  gemm_wmma_kernel<false><<<N / 32, 128, lds128, stream>>>(x, nullptr, nullptr,
      w_lin, nullptr, Wh, D_, 1.0f, 0);
  gemm_wmma_kernel<false><<<E / 16, 64, lds64, stream>>>(edge_attr, nullptr, nullptr,
      w_e, nullptr, Wem, ED_, 1.0f, 0);
  gemm_wmma_kernel<false><<<N / 32, 128, lds128, stream>>>(Wh, nullptr, nullptr,
      w_q, b_q, Q, D_, 0.25f, 0);                                  // 1/sqrt(dh) = 1/4
  gemm_wmma_kernel<false><<<N / 32, 128, lds128, stream>>>(Wh, nullptr, nullptr,
      w_v, b_v, Vv, D_, 1.0f, 0);
  gemm_wmma_kernel<true><<<EK / 32, 128, lds128, stream>>>(Wh, Wem, pair_u,
      w_k, b_k, Kv, D_, 1.0f, 0);                                  // Kv = (Wh[u]+We[e])@w_k+b_k

  // zero ctx / denom / mEnc (encoded -inf = 0x007FFFFF > 0, so 0 is identity for max)
  hipMemsetAsync(ctx, 0, ((size_t)N * D_ + 2 * (size_t)N * H_) * sizeof(float), stream);

  // Attention: scores + segment-max, exp + segment-sum, weighted context
  const int wavesPerBlk = 8;
  const int gBlocks = (G + wavesPerBlk - 1) / wavesPerBlk;
  scores_kernel<<<gBlocks, 32 * wavesPerBlk, 0, stream>>>(Q, Kv, owners, sc, mEnc, G);
  expsum_kernel<<<gBlocks, 32 * wavesPerBlk, 0, stream>>>(sc, mEnc, owners, denom, G);
  ctx_kernel<<<gBlocks, 32 * wavesPerBlk, 0, stream>>>(sc, denom, Vv, owners, pair_u, ctx, G);

  // out = relu(ctx @ w_o + b_o)
  gemm_wmma_kernel<false><<<N / 32, 128, lds128, stream>>>(ctx, nullptr, nullptr,
      w_o, b_o, (float*)d_out, D_, 1.0f, 1);
}